// BinaryVQEncoder_88905823027337
// MI455X (gfx1250) — compile-verified
//
#include <hip/hip_runtime.h>

typedef __attribute__((ext_vector_type(2))) float v2f;
typedef __attribute__((ext_vector_type(4))) float v4f;
typedef __attribute__((ext_vector_type(8))) float v8f;

#define M_DIM 8192
#define K_DIM 4096
#define N_DIM 4096

#define BM 128
#define BN 64
#define BK 32
#define LDA 36   // dword stride of As rows (pad: 16B-aligned float4 stores, conflict-free frag loads)
#define LDB 36   // dword stride of Bs rows (Bs is transposed: [n][k])

// Output layout (concatenated flat, reference return order):
//   [0]                indices  (8192*4096, written as float 0.0/1.0)
//   [M*N]              embedding (8192)
//   [M*N + 8192]       quantized_st (8192*4096)
//   [2*M*N + 8192]     z_e (8192*4096)
#define EMB_CNT 8192

__global__ __launch_bounds__(256)
void vq_gemm_fused_kernel(const float* __restrict__ x,
                          const float* __restrict__ W,
                          const float* __restrict__ bias,
                          const float* __restrict__ emb,
                          float* __restrict__ out)
{
    __shared__ float As[BM * LDA];
    __shared__ float Bs[BN * LDB];

    const int tid     = threadIdx.x;
    const int lane    = tid & 31;
    const int wave    = tid >> 5;           // 0..7, owns 16 rows of the block tile
    const int block_m = blockIdx.y * BM;
    const int block_n = blockIdx.x * BN;

    // WMMA fragment coordinates (ISA 7.12.2, f32 16x16x4):
    //  A: lane l holds row (l&15), K pair starting at 2*(l>>4)
    //  C: lane l, vgpr v holds row 8*(l>>4)+v, col (l&15)
    const int fm = lane & 15;
    const int kb = (lane >> 4) * 2;

    v8f acc[4];
#pragma unroll
    for (int j = 0; j < 4; ++j) acc[j] = (v8f)0.0f;

    // Staging index math (256 threads)
    const int a_r0 = tid >> 3;            // 0..31, 4 passes of 32 rows
    const int a_c  = (tid & 7) * 4;       // float4 column within BK=32
    const int b_r0 = tid >> 4;            // 0..15, 2 passes of 16 k-rows
    const int b_c  = (tid & 15) * 4;      // float4 column within BN=64

    for (int k0 = 0; k0 < K_DIM; k0 += BK) {
        // ---- stage A tile: BM x BK, row-major, coalesced b128 loads ----
#pragma unroll
        for (int p = 0; p < 4; ++p) {
            const int r = a_r0 + p * 32;
            v4f a = *(const v4f*)(x + (size_t)(block_m + r) * K_DIM + (k0 + a_c));
            *(v4f*)(As + r * LDA + a_c) = a;
        }
        // ---- stage B tile transposed into Bs[n][k]: coalesced global reads ----
#pragma unroll
        for (int p = 0; p < 2; ++p) {
            const int r = b_r0 + p * 16;  // k row
            v4f bv = *(const v4f*)(W + (size_t)(k0 + r) * N_DIM + (block_n + b_c));
            Bs[(b_c + 0) * LDB + r] = bv.x;
            Bs[(b_c + 1) * LDB + r] = bv.y;
            Bs[(b_c + 2) * LDB + r] = bv.z;
            Bs[(b_c + 3) * LDB + r] = bv.w;
        }
        __syncthreads();

        // ---- 8 x (4 WMMA) per K-chunk: v_wmma_f32_16x16x4_f32 ----
#pragma unroll
        for (int kk = 0; kk < BK; kk += 4) {
            v2f afrag = *(const v2f*)(As + (wave * 16 + fm) * LDA + (kk + kb));
#pragma unroll
            for (int j = 0; j < 4; ++j) {
                v2f bfrag = *(const v2f*)(Bs + (j * 16 + fm) * LDB + (kk + kb));
                acc[j] = __builtin_amdgcn_wmma_f32_16x16x4_f32(
                    /*neg_a=*/false, afrag,
                    /*neg_b=*/false, bfrag,
                    /*c_mod=*/(short)0, acc[j],
                    /*reuse_a=*/false, /*reuse_b=*/false);
            }
        }
        __syncthreads();
    }

    // ---- fused VQ epilogue ----
    const size_t MN      = (size_t)M_DIM * N_DIM;
    const size_t IDX_OFF = 0;
    const size_t Q_OFF   = MN + EMB_CNT;
    const size_t ZE_OFF  = 2 * MN + EMB_CNT;
    const int half = lane >> 4;

#pragma unroll
    for (int j = 0; j < 4; ++j) {
        const int col = block_n + j * 16 + fm;
        const float bc = bias[col];
        const float e0 = emb[col * 2 + 0];
        const float e1 = emb[col * 2 + 1];
#pragma unroll
        for (int v = 0; v < 8; ++v) {
            const int row = block_m + wave * 16 + half * 8 + v;
            const float z  = acc[j][v] + bc;
            const float d0 = (z - e0) * (z - e0);
            const float d1 = (z - e1) * (z - e1);
            const int   idx = (d1 < d0) ? 1 : 0;     // ties -> 0 (jnp.argmin picks first)
            const float q   = idx ? e1 : e0;
            const size_t off = (size_t)row * N_DIM + col;
            out[IDX_OFF + off] = (float)idx;
            out[Q_OFF  + off] = q;
            out[ZE_OFF + off] = z;
        }
    }
}

__global__ void vq_emb_copy_kernel(const float* __restrict__ emb, float* __restrict__ out)
{
    const int i = blockIdx.x * blockDim.x + threadIdx.x;
    if (i < EMB_CNT) out[(size_t)M_DIM * N_DIM + i] = emb[i];
}

extern "C" void kernel_launch(void* const* d_in, const int* in_sizes, int n_in,
                              void* d_out, int out_size, void* d_ws, size_t ws_size,
                              hipStream_t stream)
{
    const float* x    = (const float*)d_in[0];   // 8192 x 4096
    const float* W    = (const float*)d_in[1];   // 4096 x 4096
    const float* bias = (const float*)d_in[2];   // 4096
    const float* emb  = (const float*)d_in[3];   // 4096 x 2 x 1
    float* out = (float*)d_out;

    dim3 grid(N_DIM / BN, M_DIM / BM);           // (64, 64)
    vq_gemm_fused_kernel<<<grid, 256, 0, stream>>>(x, W, bias, emb, out);
    vq_emb_copy_kernel<<<(EMB_CNT + 255) / 256, 256, 0, stream>>>(emb, out);
}